// CentroidLoss_26517128085920
// MI455X (gfx1250) — compile-verified
//
#include <hip/hip_runtime.h>
#include <hip/hip_bf16.h>
#include <stdint.h>

// Problem constants (reference hardcodes D=1024; others derived from in_sizes).
#define D_DIM   1024
#define D_LOG2  10
#define WAVES_PER_BLOCK 4
#define BLOCK_THREADS   (WAVES_PER_BLOCK * 32)
#define GRID_X          256

#if defined(__gfx1250__)
#define GFX1250_ASYNC 1
#else
#define GFX1250_ASYNC 0
#endif

#if GFX1250_ASYNC
// One per-lane 16B async copy HBM -> LDS. The same signed 24-bit IOFFSET is
// added to BOTH the global and the LDS address (CDNA5 ISA),
// so a single address setup serves all 8 chunks of a 4KB row.
// th:TH_LOAD_NT: the 128MB centroid stream is single-use; keep L2 for C.
template <int OFF>
__device__ __forceinline__ void async_ld_b128(unsigned lds_addr, unsigned long long gaddr) {
  asm volatile("global_load_async_to_lds_b128 %0, %1, off offset:%c2 th:TH_LOAD_NT"
               :: "v"(lds_addr), "v"(gaddr), "n"(OFF)
               : "memory");
}
__device__ __forceinline__ void s_wait_asynccnt_8() {
  asm volatile("s_wait_asynccnt 8" ::: "memory");  // current row's 8 copies done
}
__device__ __forceinline__ void s_wait_asynccnt_0() {
  asm volatile("s_wait_asynccnt 0" ::: "memory");  // drain
}

// Issue one 4KB centroid row (1024 f32) into an LDS buffer: 8 wave-wide b128
// async copies. Caller guarantees wave-uniform control flow (EXEC all ones).
__device__ __forceinline__ void issue_row(const float* rowG, const float* rowL, int lane) {
  // Generic pointer to __shared__: low 32 bits are the wave-relative LDS
  // byte offset (aperture bits live in [63:32]) — exactly what VDST expects.
  unsigned l = (unsigned)(uintptr_t)rowL + (unsigned)lane * 16u;
  unsigned long long g = (unsigned long long)(uintptr_t)(rowG + lane * 4);
  async_ld_b128<0>(l, g);
  async_ld_b128<512>(l, g);
  async_ld_b128<1024>(l, g);
  async_ld_b128<1536>(l, g);
  async_ld_b128<2048>(l, g);
  async_ld_b128<2560>(l, g);
  async_ld_b128<3072>(l, g);
  async_ld_b128<3584>(l, g);
}
#endif  // GFX1250_ASYNC

// Sum_i |row[i] - C[units[(i*L+t)>>10], (i*L+t)&1023]| for this lane's i's.
// row points at a 1024-f32 centroid row (LDS buffer or global fallback).
__device__ __forceinline__ float row_absdiff(const float* __restrict__ row,
                                             const int* __restrict__ unitsB,
                                             const float* __restrict__ Cm,
                                             int t, int L, int lane) {
  float acc = 0.0f;
#pragma unroll
  for (int k = 0; k < 8; ++k) {
    const int idx = k * 32 + lane;          // float4 index within the row
    const float4 v = ((const float4*)row)[idx];
    int f = idx * 4 * L + t;                // flat index; max ~4.2M, fits int32
    const float vv[4] = {v.x, v.y, v.z, v.w};
#pragma unroll
    for (int j = 0; j < 4; ++j) {
      const int r = f >> D_LOG2;            // valid: r < L <= T, no clip needed
      const int c = f & (D_DIM - 1);
      const int u = unitsB[r];              // L2-resident gather (256KB)
      acc += fabsf(vv[j] - Cm[(u << D_LOG2) + c]);  // L2-resident gather (4MB)
      f += L;
    }
  }
  return acc;
}

__global__ __launch_bounds__(BLOCK_THREADS)
void centroid_loss_partial(const float* __restrict__ centroids,  // (B,T,D)
                           const int* __restrict__ units,        // (B,T)
                           const int* __restrict__ lengths,      // (B,)
                           const float* __restrict__ Cm,         // (K,D)
                           float* __restrict__ partials,         // (B, GRID_X)
                           int T) {
  __shared__ __align__(16) float lds_rows[WAVES_PER_BLOCK * 2 * D_DIM];  // 32KB
  __shared__ float wave_part[WAVES_PER_BLOCK];

  const int b    = blockIdx.y;
  const int L    = lengths[b];              // wave/block-uniform
  const int lane = threadIdx.x & 31;
  const int wave = threadIdx.x >> 5;

  const float* cenB   = centroids + (size_t)b * T * D_DIM;
  const int*   unitsB = units + (size_t)b * T;

  float* buf0 = &lds_rows[wave * 2 * D_DIM];
  float* buf1 = buf0 + D_DIM;

  const int tStride = GRID_X * WAVES_PER_BLOCK;   // 1024
  int t = blockIdx.x * WAVES_PER_BLOCK + wave;

  float acc = 0.0f;

#if GFX1250_ASYNC
  // Double-buffered async pipeline: prefetch row t+stride while consuming row t.
  int cur = 0;
  if (t < L) issue_row(cenB + (size_t)t * D_DIM, buf0, lane);
  while (t < L) {
    const int tn = t + tStride;
    const bool hasNext = (tn < L);
    if (hasNext) issue_row(cenB + (size_t)tn * D_DIM, cur ? buf0 : buf1, lane);
    if (hasNext) s_wait_asynccnt_8();   // next's 8 outstanding; current complete
    else         s_wait_asynccnt_0();
    const float* rowL = cur ? buf1 : buf0;
    acc += row_absdiff(rowL, unitsB, Cm, t, L, lane);
    cur ^= 1;
    t = tn;
  }
#else
  // Host-parse / non-gfx1250 fallback: direct b128 global loads.
  for (; t < L; t += tStride)
    acc += row_absdiff(cenB + (size_t)t * D_DIM, unitsB, Cm, t, L, lane);
#endif

  // Wave32 reduce (deterministic), then fixed-order block combine.
#pragma unroll
  for (int off = 16; off > 0; off >>= 1)
    acc += __shfl_xor(acc, off, 32);
  if (lane == 0) wave_part[wave] = acc;
  __syncthreads();
  if (threadIdx.x == 0) {
    float s = 0.0f;
#pragma unroll
    for (int w = 0; w < WAVES_PER_BLOCK; ++w) s += wave_part[w];
    partials[(size_t)b * GRID_X + blockIdx.x] = s / (float)L;   // per-sample /L
  }
}

__global__ __launch_bounds__(256)
void centroid_loss_reduce(const float* __restrict__ partials, int n, float invB,
                          float* __restrict__ out) {
  __shared__ float sm[256];
  float s = 0.0f;
  for (int i = threadIdx.x; i < n; i += 256) s += partials[i];  // fixed order
  sm[threadIdx.x] = s;
  __syncthreads();
#pragma unroll
  for (int k = 128; k > 0; k >>= 1) {
    if ((int)threadIdx.x < k) sm[threadIdx.x] += sm[threadIdx.x + k];
    __syncthreads();
  }
  if (threadIdx.x == 0) out[0] = sm[0] * invB;                  // mean over B
}

extern "C" void kernel_launch(void* const* d_in, const int* in_sizes, int n_in,
                              void* d_out, int out_size, void* d_ws, size_t ws_size,
                              hipStream_t stream) {
  const float* centroids = (const float*)d_in[0];
  const int*   units     = (const int*)d_in[1];
  const int*   lengths   = (const int*)d_in[2];
  const float* Cm        = (const float*)d_in[3];

  const int B = in_sizes[2];
  const int T = in_sizes[1] / B;

  float* partials = (float*)d_ws;          // B * GRID_X floats (fully rewritten)

  dim3 grid(GRID_X, B);
  centroid_loss_partial<<<grid, BLOCK_THREADS, 0, stream>>>(
      centroids, units, lengths, Cm, partials, T);
  centroid_loss_reduce<<<1, 256, 0, stream>>>(
      partials, GRID_X * B, 1.0f / (float)B, (float*)d_out);
}